// SelfConsensus_61813169324646
// MI455X (gfx1250) — compile-verified
//
#include <hip/hip_runtime.h>
#include <hip/hip_bf16.h>

typedef float v2f __attribute__((ext_vector_type(2)));
typedef float v8f __attribute__((ext_vector_type(8)));

constexpr int B_   = 2;
constexpr int L_   = 2048;
constexpr int DIM_ = 512;
constexpr int H_   = 8;
constexpr int HD_  = 64;
constexpr int EH_  = 16;
constexpr int T_   = 2;
constexpr int R_   = 8;

__device__ __forceinline__ float gelu_f(float v) {
    return 0.5f * v * (1.0f + erff(v * 0.70710678118654752f));
}
__device__ __forceinline__ float softplus_f(float v) {
    return (v > 20.0f) ? v : log1pf(__expf(v));
}
__device__ __forceinline__ v8f zero8() {
    v8f z = {0.f, 0.f, 0.f, 0.f, 0.f, 0.f, 0.f, 0.f};
    return z;
}

// ---------------------------------------------------------------------------
// C[M,512] = A[M,512] @ Bw[512,512] + bias  via fp32 WMMA 16x16x4.
// grid=(M/16, 4), block=256 (8 waves, each a 16x16 tile; shared LDS A-tile).
// ---------------------------------------------------------------------------
__global__ __launch_bounds__(256) void gemm512_wmma(
    const float* __restrict__ A, const float* __restrict__ Bw,
    const float* __restrict__ bias, float* __restrict__ C)
{
    __shared__ float As[16][516];
    const int m0 = blockIdx.x << 4;
    for (int idx = threadIdx.x; idx < 16 * 512; idx += 256) {
        int r = idx >> 9, k = idx & 511;
        As[r][k] = A[(size_t)(m0 + r) * DIM_ + k];
    }
    __syncthreads();

    const int wave  = threadIdx.x >> 5;
    const int lane  = threadIdx.x & 31;
    const int n0    = (blockIdx.y << 7) + (wave << 4);
    const int col   = lane & 15;
    const int khalf = (lane >> 4) << 1;          // lanes 16-31 carry K+2,K+3
    const float* arow = As[lane & 15];

    v8f acc = zero8();
    #pragma unroll 4
    for (int k0 = 0; k0 < 512; k0 += 4) {
        const int kk = k0 + khalf;
        v2f a; a.x = arow[kk]; a.y = arow[kk + 1];
        v2f b; b.x = Bw[(size_t)kk * DIM_ + n0 + col];
               b.y = Bw[(size_t)(kk + 1) * DIM_ + n0 + col];
        acc = __builtin_amdgcn_wmma_f32_16x16x4_f32(
            false, a, false, b, (short)0, acc, false, false);
    }
    const float bv = bias[n0 + col];
    const int rbase = m0 + ((lane >> 4) << 3);
    #pragma unroll
    for (int v = 0; v < 8; ++v)
        C[(size_t)(rbase + v) * DIM_ + n0 + col] = acc[v] + bv;
}

// ---------------------------------------------------------------------------
// Per-edge MLPs: wave = 16 edges. ef (K=1024) staged via LDS in 16x64 chunks,
// one A-frag feeds both a1 (attention) and l1 (Lam) WMMAs.
// Outputs: alphas[b,e] = softplus(gelu(ef@a1+b)@a2+b), h2[b,e,16] = gelu(ef@l1+b)
// ---------------------------------------------------------------------------
__global__ __launch_bounds__(256) void edge_mlp_kernel(
    const float* __restrict__ x, const int* __restrict__ ei, const int* __restrict__ ej,
    const float* __restrict__ a1w, const float* __restrict__ a1b,
    const float* __restrict__ a2w, const float* __restrict__ a2b,
    const float* __restrict__ l1w, const float* __restrict__ l1b,
    float* __restrict__ alphas, float* __restrict__ h2out, int E)
{
    __shared__ float efs[8][16][68];
    __shared__ float hred[8][16][17];
    const int wave = threadIdx.x >> 5, lane = threadIdx.x & 31;
    const int ntiles = (E + 15) >> 4;
    const int nwaves = B_ * ntiles;
    const int gw  = blockIdx.x * 8 + wave;
    const bool wv = gw < nwaves;
    const int gwc = wv ? gw : (nwaves - 1);
    const int b   = gwc / ntiles;
    const int e0  = (gwc % ntiles) << 4;

    const int arow  = lane & 15;
    const int col   = lane & 15;
    const int khalf = (lane >> 4) << 1;
    const int vrow0 = (lane >> 4) << 3;

    v8f acc_a = zero8();
    v8f acc_l = zero8();

    for (int cc = 0; cc < 16; ++cc) {          // chunks 0-7: x[edge_i], 8-15: x[edge_j]
        __syncthreads();
        for (int sub = 0; sub < 32; ++sub) {   // coalesced stage of 16x64 chunk
            int idx = (sub << 5) + lane;
            int r = idx >> 6, kk = idx & 63;
            int e = e0 + r;
            int ec = min(e, E - 1);
            int src = (cc < 8) ? ei[ec] : ej[ec];
            int kg  = ((cc & 7) << 6) + kk;
            float v = x[((size_t)b * L_ + src) * DIM_ + kg];
            efs[wave][r][kk] = (e < E) ? v : 0.0f;
        }
        __syncthreads();
        #pragma unroll
        for (int ks = 0; ks < 16; ++ks) {
            const int kl = (ks << 2) + khalf;
            v2f a; a.x = efs[wave][arow][kl]; a.y = efs[wave][arow][kl + 1];
            const int kgl = (cc << 6) + kl;    // global k in [0,1024)
            v2f b1; b1.x = a1w[(size_t)kgl * EH_ + col];
                    b1.y = a1w[(size_t)(kgl + 1) * EH_ + col];
            acc_a = __builtin_amdgcn_wmma_f32_16x16x4_f32(
                false, a, false, b1, (short)0, acc_a, false, false);
            v2f b2; b2.x = l1w[(size_t)kgl * EH_ + col];
                    b2.y = l1w[(size_t)(kgl + 1) * EH_ + col];
            acc_l = __builtin_amdgcn_wmma_f32_16x16x4_f32(
                false, a, false, b2, (short)0, acc_l, false, false);
        }
    }

    const float ba = a1b[col], bl = l1b[col];
    #pragma unroll
    for (int v = 0; v < 8; ++v) {
        const int rv = vrow0 + v;
        hred[wave][rv][col] = gelu_f(acc_a[v] + ba);
        const int e = e0 + rv;
        if (wv && e < E)
            h2out[((size_t)b * E + e) * EH_ + col] = gelu_f(acc_l[v] + bl);
    }
    __syncthreads();
    if (lane < 16) {
        const int e = e0 + lane;
        if (wv && e < E) {
            float s = 0.f;
            #pragma unroll
            for (int c2 = 0; c2 < 16; ++c2) s += hred[wave][lane][c2] * a2w[c2];
            alphas[(size_t)b * E + e] = softplus_f(s + a2b[0]);
        }
    }
}

__global__ void zero_kernel(float* __restrict__ p, int n) {
    int i = blockIdx.x * 256 + threadIdx.x;
    if (i < n) p[i] = 0.f;
}

// ---------------------------------------------------------------------------
// One evolution step: wave = (b,h,16-edge tile). RoPE'd diff staged in LDS,
// Lam recomputed via WMMA (h2[16x16] @ l2_w[16x512] head slice, K=16),
// normalization fused as res += (lam.diff / max(|lam|^2,1e-24)) * lam,
// row reductions via shfl_xor butterflies, fp32 atomic scatter into upd.
// ---------------------------------------------------------------------------
__global__ __launch_bounds__(256) void edge_update_kernel(
    const float* __restrict__ u, float* __restrict__ upd,
    const float* __restrict__ h2, const float* __restrict__ alphas,
    const float* __restrict__ l2w, const float* __restrict__ l2b,
    const int* __restrict__ ei, const int* __restrict__ ej, int E)
{
    __shared__ float sdiff[8][16][68];
    const int wave = threadIdx.x >> 5, lane = threadIdx.x & 31;
    const int ntiles = (E + 15) >> 4;
    const int total  = ntiles << 4;            // ntiles * B_*H_
    const int gw  = blockIdx.x * 8 + wave;
    const bool wv = gw < total;
    const int gwc = wv ? gw : (total - 1);
    const int e0  = (gwc >> 4) << 4;
    const int bh  = gwc & 15;
    const int b   = bh >> 3, h = bh & 7;

    for (int sub = 0; sub < 32; ++sub) {       // diff = RoPE(u_i, i-j) - u_j
        int idx = (sub << 5) + lane;
        int r = idx >> 6, d = idx & 63;
        int e = e0 + r, ec = min(e, E - 1);
        int pi = ei[ec], pj = ej[ec];
        size_t bi = ((size_t)b * L_ + pi) * DIM_ + h * HD_;
        size_t bj = ((size_t)b * L_ + pj) * DIM_ + h * HD_;
        float uiv = u[bi + d];
        int dr = (d < 32) ? (d + 32) : (d - 32);
        float uo  = u[bi + dr];
        float rot = (d < 32) ? -uo : uo;
        int dp = d & 31;
        float invf = __expf((float)(2 * dp) * (-9.210340371976184f / 64.0f));
        float ang  = (float)(pi - pj) * invf;
        float uir  = uiv * __cosf(ang) + rot * __sinf(ang);
        float dv   = uir - u[bj + d];
        sdiff[wave][r][d] = (e < E) ? dv : 0.f;
    }
    __syncthreads();

    const int col   = lane & 15;
    const int khalf = (lane >> 4) << 1;
    const int vrow0 = (lane >> 4) << 3;

    float ah[8];                               // h2 A-frags (4 k-steps of K=16)
    {
        const int eA = e0 + (lane & 15);
        const int ec = min(eA, E - 1);
        const float* hr = &h2[((size_t)b * E + ec) * EH_];
        const bool av = eA < E;
        #pragma unroll
        for (int ks = 0; ks < 4; ++ks) {
            ah[2 * ks]     = av ? hr[4 * ks + khalf]     : 0.f;
            ah[2 * ks + 1] = av ? hr[4 * ks + khalf + 1] : 0.f;
        }
    }
    float alph[8];
    #pragma unroll
    for (int v = 0; v < 8; ++v) {
        const int e = e0 + vrow0 + v;
        alph[v] = (e < E) ? alphas[(size_t)b * E + e] : 0.f;
    }
    v8f res[4];
    #pragma unroll
    for (int ct = 0; ct < 4; ++ct)
        #pragma unroll
        for (int v = 0; v < 8; ++v)
            res[ct][v] = alph[v] * sdiff[wave][vrow0 + v][ct * 16 + col];

    const int cbase = h * (R_ * HD_);          // per-head column base in l2_w output
    for (int r = 0; r < 8; ++r) {
        v8f lam[4];
        #pragma unroll
        for (int ct = 0; ct < 4; ++ct) {
            v8f acc = zero8();
            const int cg = cbase + r * 64 + ct * 16 + col;
            #pragma unroll
            for (int ks = 0; ks < 4; ++ks) {
                v2f a; a.x = ah[2 * ks]; a.y = ah[2 * ks + 1];
                const int kk = 4 * ks + khalf;
                v2f bb; bb.x = l2w[(size_t)kk * 4096 + cg];
                        bb.y = l2w[(size_t)(kk + 1) * 4096 + cg];
                acc = __builtin_amdgcn_wmma_f32_16x16x4_f32(
                    false, a, false, bb, (short)0, acc, false, false);
            }
            const float bv = l2b[cg];
            #pragma unroll
            for (int v = 0; v < 8; ++v) acc[v] += bv;
            lam[ct] = acc;
        }
        float n2[8], sd[8];
        #pragma unroll
        for (int v = 0; v < 8; ++v) {
            float p = 0.f, q = 0.f;
            #pragma unroll
            for (int ct = 0; ct < 4; ++ct) {
                float lv = lam[ct][v];
                p += lv * lv;
                q += lv * sdiff[wave][vrow0 + v][ct * 16 + col];
            }
            n2[v] = p; sd[v] = q;
        }
        #pragma unroll
        for (int m = 1; m < 16; m <<= 1) {
            #pragma unroll
            for (int v = 0; v < 8; ++v) {
                n2[v] += __shfl_xor(n2[v], m, 32);
                sd[v] += __shfl_xor(sd[v], m, 32);
            }
        }
        #pragma unroll
        for (int v = 0; v < 8; ++v) {
            const float sc = sd[v] / fmaxf(n2[v], 1e-24f);
            #pragma unroll
            for (int ct = 0; ct < 4; ++ct) res[ct][v] += sc * lam[ct][v];
        }
    }

    #pragma unroll
    for (int v = 0; v < 8; ++v) {
        const int e = e0 + vrow0 + v;
        if (wv && e < E) {
            const int pi = ei[e], pj = ej[e];
            size_t bi = ((size_t)b * L_ + pi) * DIM_ + h * HD_;
            size_t bj = ((size_t)b * L_ + pj) * DIM_ + h * HD_;
            #pragma unroll
            for (int ct = 0; ct < 4; ++ct) {
                const int d = ct * 16 + col;
                const float rv2 = res[ct][v];
                unsafeAtomicAdd(&upd[bi + d], rv2);
                unsafeAtomicAdd(&upd[bj + d], -rv2);
            }
        }
    }
}

// u -= softplus(step_sizes[t, l]) * upd ; upd = 0 (ready for next step/replay)
__global__ void apply_update_kernel(float* __restrict__ u, float* __restrict__ upd,
                                    const float* __restrict__ ss, int t, int n)
{
    int i = blockIdx.x * 256 + threadIdx.x;
    if (i < n) {
        const int l = (i >> 9) & (L_ - 1);     // DIM_=512, L_ power of two
        const float eta = softplus_f(ss[t * L_ + l]);
        u[i] -= eta * upd[i];
        upd[i] = 0.f;
    }
}

extern "C" void kernel_launch(void* const* d_in, const int* in_sizes, int n_in,
                              void* d_out, int out_size, void* d_ws, size_t ws_size,
                              hipStream_t stream) {
    const float* x    = (const float*)d_in[0];
    const int*   ei   = (const int*)  d_in[1];
    const int*   ej   = (const int*)  d_in[2];
    const float* te_w = (const float*)d_in[3];
    const float* te_b = (const float*)d_in[4];
    const float* a1w  = (const float*)d_in[5];
    const float* a1b  = (const float*)d_in[6];
    const float* a2w  = (const float*)d_in[7];
    const float* a2b  = (const float*)d_in[8];
    const float* l1w  = (const float*)d_in[9];
    const float* l1b  = (const float*)d_in[10];
    const float* l2w  = (const float*)d_in[11];
    const float* l2b  = (const float*)d_in[12];
    const float* ss   = (const float*)d_in[13];
    const float* ow   = (const float*)d_in[14];
    const float* ob   = (const float*)d_in[15];
    float* out = (float*)d_out;
    const int E = in_sizes[1];

    float* ws = (float*)d_ws;
    const size_t UN = (size_t)B_ * L_ * DIM_;
    float* u      = ws;
    float* upd    = ws + UN;
    float* alphas = ws + 2 * UN;
    float* h2     = alphas + (size_t)B_ * E;

    const int M = B_ * L_;
    gemm512_wmma<<<dim3(M / 16, DIM_ / 128), 256, 0, stream>>>(x, te_w, te_b, u);

    const int ntiles = (E + 15) / 16;
    const int w2 = B_ * ntiles;
    edge_mlp_kernel<<<(w2 + 7) / 8, 256, 0, stream>>>(
        x, ei, ej, a1w, a1b, a2w, a2b, l1w, l1b, alphas, h2, E);

    const int n = B_ * L_ * DIM_;
    zero_kernel<<<(n + 255) / 256, 256, 0, stream>>>(upd, n);

    const int w3 = ntiles * (B_ * H_);
    for (int t = 0; t < T_; ++t) {
        edge_update_kernel<<<(w3 + 7) / 8, 256, 0, stream>>>(
            u, upd, h2, alphas, l2w, l2b, ei, ej, E);
        apply_update_kernel<<<(n + 255) / 256, 256, 0, stream>>>(u, upd, ss, t, n);
    }

    gemm512_wmma<<<dim3(M / 16, DIM_ / 128), 256, 0, stream>>>(u, ow, ob, out);
}